// VocabParallelEmbeddingWithTopping_63814624084196
// MI455X (gfx1250) — compile-verified
//
#include <hip/hip_runtime.h>
#include <stdint.h>

// ---------------------------------------------------------------------------
// VocabParallelEmbeddingWithTopping: routed embedding gather.
//   out[t, :] = (w[t] == -1) ? base[id[t], :] : delta[w[t], id[t], :]
// Memory-bound: ~64 MB total traffic -> ~2.8 us at 23.3 TB/s. No matrix math,
// so the CDNA5 feature of interest is the async global<->LDS DMA path
// (ASYNCcnt, global_load_async_to_lds_b128 / global_store_async_from_lds_b128)
// with b128 transfers, not WMMA.
// ---------------------------------------------------------------------------

#define EMBED_DIM        1024
#define ROW_BYTES        (EMBED_DIM * 4)          // 4096 B per row
#define THREADS          256                      // 8 wave32 waves
#define TOKENS_PER_BLOCK 4                        // 16 KB LDS staging

// Exact pointee type from the builtin signature: vector_size(16) int.
typedef int v4i __attribute__((vector_size(16)));
typedef __attribute__((address_space(1))) v4i g_v4i;   // global (__device__)
typedef __attribute__((address_space(3))) v4i l_v4i;   // LDS   (__shared__)

#if __has_builtin(__builtin_amdgcn_global_load_async_to_lds_b128) && \
    __has_builtin(__builtin_amdgcn_global_store_async_from_lds_b128)
#define USE_ASYNC_LDS 1
#else
#define USE_ASYNC_LDS 0
#endif

__device__ __forceinline__ void wait_asynccnt0() {
#if __has_builtin(__builtin_amdgcn_s_wait_asynccnt)
    __builtin_amdgcn_s_wait_asynccnt(0);
#else
    asm volatile("s_wait_asynccnt 0" ::: "memory");
#endif
}

__global__ __launch_bounds__(THREADS)
void emb_route_gather_kernel(const int* __restrict__ input_ids,
                             const int* __restrict__ weight_indices,
                             const float* __restrict__ base_w,
                             const float* __restrict__ delta_w,
                             float* __restrict__ out,
                             int vocab, int n_tokens)
{
    const int tid   = threadIdx.x;
    const int tok0  = blockIdx.x * TOKENS_PER_BLOCK;
    const unsigned off = (unsigned)tid * 16u;     // this lane's 16 B chunk
    // EMBED_DIM*4 / (THREADS*16) == 1 chunk per lane per row.

#if USE_ASYNC_LDS
    __shared__ char smem[TOKENS_PER_BLOCK * ROW_BYTES];

    // Phase 1: fire all async global->LDS b128 copies, one row per token.
    // Uniform per-token index loads scalarize to s_load.
    for (int k = 0; k < TOKENS_PER_BLOCK; ++k) {
        const int t = tok0 + k;
        if (t >= n_tokens) break;
        const int    id = input_ids[t];
        const int    w  = weight_indices[t];
        const float* src = (w < 0)
            ? base_w  + (size_t)id * EMBED_DIM
            : delta_w + ((size_t)w * (size_t)vocab + (size_t)id) * EMBED_DIM;
        __builtin_amdgcn_global_load_async_to_lds_b128(
            (g_v4i*)((const char*)src + off),
            (l_v4i*)(smem + k * ROW_BYTES + off),
            /*imm offset*/0, /*cpol*/0);
    }

    // Each lane's LDS bytes are private to that lane, so a single ASYNCcnt
    // drain orders its loads before its stores; no workgroup barrier needed.
    wait_asynccnt0();

    // Phase 2: async LDS->global b128 stores into the output rows.
    for (int k = 0; k < TOKENS_PER_BLOCK; ++k) {
        const int t = tok0 + k;
        if (t >= n_tokens) break;
        float* dst = out + (size_t)t * EMBED_DIM;
        __builtin_amdgcn_global_store_async_from_lds_b128(
            (g_v4i*)((char*)dst + off),
            (l_v4i*)(smem + k * ROW_BYTES + off),
            /*imm offset*/0, /*cpol*/0);
    }
    // S_ENDPGM performs an implicit wait-idle, draining outstanding stores.
#else
    // Fallback: straight VGPR float4 copy (still b128 traffic).
    for (int k = 0; k < TOKENS_PER_BLOCK; ++k) {
        const int t = tok0 + k;
        if (t >= n_tokens) break;
        const int id = input_ids[t];
        const int w  = weight_indices[t];
        const float* src = (w < 0)
            ? base_w  + (size_t)id * EMBED_DIM
            : delta_w + ((size_t)w * (size_t)vocab + (size_t)id) * EMBED_DIM;
        float* dst = out + (size_t)t * EMBED_DIM;
        const float4 v = *(const float4*)((const char*)src + off);
        *(float4*)((char*)dst + off) = v;
    }
#endif
}

extern "C" void kernel_launch(void* const* d_in, const int* in_sizes, int n_in,
                              void* d_out, int out_size, void* d_ws, size_t ws_size,
                              hipStream_t stream) {
    (void)n_in; (void)d_ws; (void)ws_size;
    const int*   input_ids      = (const int*)d_in[0];
    const int*   weight_indices = (const int*)d_in[1];
    const float* base_w         = (const float*)d_in[2];
    const float* delta_w        = (const float*)d_in[3];
    float*       out            = (float*)d_out;

    const int n_tokens = in_sizes[0];                 // 8192
    const int vocab    = in_sizes[2] / EMBED_DIM;     // 32000
    (void)out_size;

    const int grid = (n_tokens + TOKENS_PER_BLOCK - 1) / TOKENS_PER_BLOCK;
    emb_route_gather_kernel<<<grid, THREADS, 0, stream>>>(
        input_ids, weight_indices, base_w, delta_w, out, vocab, n_tokens);
}